// MultiVariationalDist_41781441855780
// MI455X (gfx1250) — compile-verified
//
#include <hip/hip_runtime.h>

// B=64, NTH=56, NTW=56 -> T = 200704 tiles; M=8 slots; DF=8 fluxes; out row = 11 floats
#define NTILES   200704
#define MM       8
#define DFF      8
#define DOUTW    11
#define FARV     100.0f

typedef __attribute__((ext_vector_type(2))) float v2f;
typedef __attribute__((ext_vector_type(8))) float v8f;

// One wave handles 2 consecutive tiles.
// dist^2(r,c) = ex*(-2tx) + ey*(-2ty) + |e|^2*1 + 1*|t|^2  -> K=4 WMMA.
// A 16x4 f32 layout: lanes 0-15 hold {K0,K1} of row (lane&15); lanes 16-31 hold {K2,K3}.
// B 4x16 f32 layout: lanes 0-15 hold {K0,K1} of col (lane&15); lanes 16-31 hold {K2,K3}.
// D 16x16 f32: lane L<16: c[v] = D[v][L]; lane L>=16: c[v] = D[v+8][L-16].
__global__ __launch_bounds__(256) void mvd_match_kernel(
    const float* __restrict__ est_locs,    // [T,8,2]
    const float* __restrict__ true_locs,   // [T,8,2]
    const float* __restrict__ true_fluxes, // [T,8,8]
    const int*   __restrict__ est_n,       // [T]
    const int*   __restrict__ true_n,      // [T]
    float* __restrict__ out)               // [T,8,11]
{
    __shared__ float buf[8][2 * MM * DOUTW];   // 176 floats per wave

    const int lane  = threadIdx.x & 31;
    const int wslot = threadIdx.x >> 5;
    const int wave  = blockIdx.x * (blockDim.x >> 5) + wslot;
    const int t0    = wave * 2;                // first of this wave's two tiles

    // Early prefetch of this wave's flux tiles (global_prefetch_b8 path).
    __builtin_prefetch(&true_fluxes[(size_t)t0 * (MM * DFF)], 0, 1);

    const int rowA  = lane & 15;               // global WMMA row/col 0..15
    const int tA    = rowA >> 3;               // 0 -> tile t0, 1 -> tile t0+1
    const int slotA = rowA & 7;
    const int hi    = lane >> 4;               // which K-half this lane supplies
    const int tile  = t0 + tA;

    const int ec = est_n[tile];
    const int tc = true_n[tile];

    v2f e = *(const v2f*)&est_locs[(size_t)tile * (MM * 2) + slotA * 2];
    v2f t = *(const v2f*)&true_locs[(size_t)tile * (MM * 2) + slotA * 2];
    const float ex = (slotA < ec) ? e.x : FARV;
    const float ey = (slotA < ec) ? e.y : FARV;
    const float tx = (slotA < tc) ? t.x : FARV;
    const float ty = (slotA < tc) ? t.y : FARV;
    const float en2 = ex * ex + ey * ey;
    const float tn2 = tx * tx + ty * ty;

    v2f a1, b1, a2, b2;
    if (hi == 0) {
        a1.x = ex;         a1.y = ey;          // A rows = est:  {ex, ey, |e|^2, 1}
        b1.x = -2.0f * tx; b1.y = -2.0f * ty;  // B cols = true: {-2tx,-2ty, 1, |t|^2}
        a2.x = tx;         a2.y = ty;          // transposed problem
        b2.x = -2.0f * ex; b2.y = -2.0f * ey;
    } else {
        a1.x = en2;  a1.y = 1.0f;
        b1.x = 1.0f; b1.y = tn2;
        a2.x = tn2;  a2.y = 1.0f;
        b2.x = 1.0f; b2.y = en2;
    }

    v8f c1 = {0.0f, 0.0f, 0.0f, 0.0f, 0.0f, 0.0f, 0.0f, 0.0f};
    v8f c2 = {0.0f, 0.0f, 0.0f, 0.0f, 0.0f, 0.0f, 0.0f, 0.0f};
    // D = dist^2 (rows=est, cols=true) ; D' = transpose. EXEC is all-ones here.
    c1 = __builtin_amdgcn_wmma_f32_16x16x4_f32(false, a1, false, b1, (short)0, c1, false, false);
    c2 = __builtin_amdgcn_wmma_f32_16x16x4_f32(false, a2, false, b2, (short)0, c2, false, false);

    // Per-lane argmins (first-index tie-break, matches jnp.argmin; sqrt omitted: monotone).
    // Lane j in 0..7  : c1[v] = tile0 dist2[v][j] -> match1[j]; c2[v] = tile0 dist2[j][v] -> match2[j]
    // Lane j in 24..31: same for tile1 (slot j-24).
    float bm1 = c1[0]; int m1 = 0;
    float bm2 = c2[0]; int m2 = 0;
#pragma unroll
    for (int v = 1; v < 8; ++v) {
        if (c1[v] < bm1) { bm1 = c1[v]; m1 = v; }
        if (c2[v] < bm2) { bm2 = c2[v]; m2 = v; }
    }

    const bool owner = (lane < 8) || (lane >= 24);
    const int  base  = (lane < 8) ? 0 : 24;    // lanes 8..23 get a valid-but-unused base
    const int  j     = lane - base;

    int pack = m1 | (m2 << 8);
    // one2one[j] = (match1[match2[j]] == j) && match1[j] < est_count && match2[j] < true_count
    const int m1c2 = __shfl(pack, base + m2, 32) & 0xff;
    const int one  = (m1c2 == j) && (m1 < ec) && (m2 < tc);
    pack |= (one << 16);

    // Invert the scatter: lane e owns output row e; find j with one2one && match1[j]==e.
    int found = -1;
#pragma unroll
    for (int jj = 0; jj < 8; ++jj) {
        const int pj = __shfl(pack, base + jj, 32);
        if (((pj >> 16) & 1) && ((pj & 0xff) == j)) found = (pj >> 8) & 0xff;
    }

    if (owner) {
        float o[DOUTW];
#pragma unroll
        for (int k = 0; k < DOUTW; ++k) o[k] = 0.0f;
        if (found >= 0) {
            const float* tl = &true_locs[(size_t)tile * (MM * 2) + found * 2];
            o[0] = tl[0];
            o[1] = tl[1];
            const float* fl = &true_fluxes[(size_t)tile * (MM * DFF) + found * DFF];
#pragma unroll
            for (int k = 0; k < DFF; ++k) o[2 + k] = fl[k];
            o[10] = (found < tc) ? 1.0f : 0.0f;
        }
        float* bp = &buf[wslot][tA * (MM * DOUTW) + j * DOUTW];
#pragma unroll
        for (int k = 0; k < DOUTW; ++k) bp[k] = o[k];
    }
    // Same-wave LDS store->load is kept in order by DScnt; redistribute for coalesced stores.
    float* gout = out + (size_t)t0 * (MM * DOUTW);  // 176 contiguous floats for this wave
#pragma unroll
    for (int it = 0; it < 6; ++it) {
        const int i = lane + it * 32;
        if (i < 2 * MM * DOUTW) gout[i] = buf[wslot][i];
    }
}

extern "C" void kernel_launch(void* const* d_in, const int* in_sizes, int n_in,
                              void* d_out, int out_size, void* d_ws, size_t ws_size,
                              hipStream_t stream) {
    (void)in_sizes; (void)n_in; (void)d_ws; (void)ws_size; (void)out_size;
    const float* est_locs    = (const float*)d_in[0];
    const float* true_locs   = (const float*)d_in[1];
    const float* true_fluxes = (const float*)d_in[2];
    const int*   est_n       = (const int*)d_in[3];
    const int*   true_n      = (const int*)d_in[4];
    float* out = (float*)d_out;

    // 2 tiles per wave, 8 waves per block -> 16 tiles per block; 200704/16 = 12544 blocks exactly.
    const int blocks = NTILES / 16;
    mvd_match_kernel<<<blocks, 256, 0, stream>>>(est_locs, true_locs, true_fluxes,
                                                 est_n, true_n, out);
}